// TeleChat2Attention_68315749810677
// MI455X (gfx1250) — compile-verified
//
#include <hip/hip_runtime.h>
#include <hip/hip_bf16.h>

typedef __attribute__((ext_vector_type(2))) float v2f;
typedef __attribute__((ext_vector_type(8))) float v8f;

constexpr int kB   = 2;
constexpr int kS   = 2048;
constexpr int kHID = 4096;
constexpr int kNH  = 32;
constexpr int kNKV = 8;
constexpr int kHD  = 128;

// LDS row stride (floats): 36*4 = 144B keeps 16B alignment for async b128
// stores and gives 16 distinct banks across a 16-lane column read group.
constexpr int kLdsStride = 36;

// ---------------------------------------------------------------------------
// Issue a 64x32-float tile load, global -> LDS, via async DMA (ASYNCcnt).
// 128 threads x 4 issues x 16B = 8KB tile. Per-lane LDS dest address is the
// low 32 bits of the generic pointer (flat addr [31:0] is the LDS offset).
// ---------------------------------------------------------------------------
__device__ __forceinline__ void async_tile_load(
    const float* __restrict__ base, int row0, int K, int kc,
    float (*lds)[kLdsStride], int tid)
{
#pragma unroll
  for (int i = 0; i < 4; ++i) {
    int idx = tid + i * 128;
    int r   = idx >> 3;         // 0..63
    int c4  = (idx & 7) << 2;   // 0,4,...,28
    unsigned lds_addr = (unsigned)(size_t)&lds[r][c4];
    const float* gptr = base + (size_t)(row0 + r) * K + kc + c4;
    asm volatile("global_load_async_to_lds_b128 %0, %1, off"
                 :: "v"(lds_addr), "v"(gptr) : "memory");
  }
}

#define WAIT_ASYNCCNT_8() asm volatile("s_wait_asynccnt 0x8" ::: "memory")
#define WAIT_ASYNCCNT_0() asm volatile("s_wait_asynccnt 0x0" ::: "memory")

// ---------------------------------------------------------------------------
// Generic fp32 WMMA GEMM: C[M,N] = A[M,K] @ B[N,K]^T (+bias), row-major.
// Block tile 64x64, 4 waves (wave32) in 2x2, each wave 32x32 = 2x2 WMMA tiles.
// Double-buffered LDS fed by global_load_async_to_lds_b128.
// ---------------------------------------------------------------------------
__global__ __launch_bounds__(128) void wmma_gemm_f32(
    const float* __restrict__ A, const float* __restrict__ Bw,
    const float* __restrict__ bias, float* __restrict__ C,
    int M, int N, int K)
{
  __shared__ __align__(16) float lA[2][64][kLdsStride];
  __shared__ __align__(16) float lB[2][64][kLdsStride];

  const int tid  = threadIdx.x;
  const int lane = tid & 31;
  const int wave = tid >> 5;
  const int half = lane >> 4;     // 0: lanes 0-15, 1: lanes 16-31
  const int klo  = half << 1;     // K sub-offset per ISA 32-bit A/B layout
  const int col  = lane & 15;
  const int m0   = blockIdx.y * 64;
  const int n0   = blockIdx.x * 64;
  const int wm   = (wave >> 1) * 32;
  const int wn   = (wave & 1) * 32;

  v8f acc[2][2] = {};

  // Prologue: start streaming chunk 0 into buffer 0.
  async_tile_load(A,  m0, K, 0, lA[0], tid);
  async_tile_load(Bw, n0, K, 0, lB[0], tid);

  for (int kc = 0; kc < K; kc += 32) {
    const int cur = (kc >> 5) & 1;

    if (kc + 32 < K) {
      // Stream next chunk into the other buffer; it overlaps this chunk's
      // WMMAs. Then wait for the *previous* 8 issues (this chunk's data).
      async_tile_load(A,  m0, K, kc + 32, lA[cur ^ 1], tid);
      async_tile_load(Bw, n0, K, kc + 32, lB[cur ^ 1], tid);
      WAIT_ASYNCCNT_8();
    } else {
      WAIT_ASYNCCNT_0();
    }
    __syncthreads();   // all waves' async data for `cur` has landed

    const float (*tA)[kLdsStride] = lA[cur];
    const float (*tB)[kLdsStride] = lB[cur];
#pragma unroll
    for (int kb = 0; kb < 32; kb += 4) {
      v2f a0, a1, b0, b1;
      a0.x = tA[wm + col][kb + klo];      a0.y = tA[wm + col][kb + klo + 1];
      a1.x = tA[wm + 16 + col][kb + klo]; a1.y = tA[wm + 16 + col][kb + klo + 1];
      b0.x = tB[wn + col][kb + klo];      b0.y = tB[wn + col][kb + klo + 1];
      b1.x = tB[wn + 16 + col][kb + klo]; b1.y = tB[wn + 16 + col][kb + klo + 1];
      acc[0][0] = __builtin_amdgcn_wmma_f32_16x16x4_f32(false, a0, false, b0, (short)0, acc[0][0], false, false);
      acc[0][1] = __builtin_amdgcn_wmma_f32_16x16x4_f32(false, a0, false, b1, (short)0, acc[0][1], false, false);
      acc[1][0] = __builtin_amdgcn_wmma_f32_16x16x4_f32(false, a1, false, b0, (short)0, acc[1][0], false, false);
      acc[1][1] = __builtin_amdgcn_wmma_f32_16x16x4_f32(false, a1, false, b1, (short)0, acc[1][1], false, false);
    }
    __syncthreads();   // nobody may overwrite `cur` until all reads are done
  }

  // Epilogue per ISA C/D layout: VGPR i -> row (i + 8*half), lane col = lane%16
#pragma unroll
  for (int mi = 0; mi < 2; ++mi)
#pragma unroll
    for (int ni = 0; ni < 2; ++ni)
#pragma unroll
      for (int i = 0; i < 8; ++i) {
        int r = m0 + wm + mi * 16 + i + 8 * half;
        int c = n0 + wn + ni * 16 + col;
        float v = acc[mi][ni][i];
        if (bias) v += bias[c];
        C[(size_t)r * N + c] = v;
      }
}

// ---------------------------------------------------------------------------
// RoPE + KV split + transpose to (B, H, S, HD)
// ---------------------------------------------------------------------------
__global__ __launch_bounds__(256) void rope_split(
    const float* __restrict__ qproj, const float* __restrict__ kvproj,
    const float* __restrict__ cosT, const float* __restrict__ sinT,
    float* __restrict__ Q, float* __restrict__ Kd, float* __restrict__ Vd)
{
  const int bs = blockIdx.x;        // b*S + s
  const int b  = bs / kS;
  const int s  = bs % kS;
  const float* cr = cosT + (size_t)s * kHD;
  const float* sr = sinT + (size_t)s * kHD;
  const float* qrow  = qproj  + (size_t)bs * kHID;
  const float* kvrow = kvproj + (size_t)bs * (2 * kNKV * kHD);

  for (int i = threadIdx.x; i < kHID; i += 256) {
    int h = i >> 7, d = i & 127;
    float x = qrow[i];
    float o = (d < 64) ? -qrow[h * kHD + d + 64] : qrow[h * kHD + d - 64];
    Q[(((size_t)b * kNH + h) * kS + s) * kHD + d] = x * cr[d] + o * sr[d];
  }
  for (int i = threadIdx.x; i < kNKV * kHD; i += 256) {
    int g = i >> 7, d = i & 127;
    const float* kb = kvrow + g * (2 * kHD);
    float x = kb[d];
    float o = (d < 64) ? -kb[d + 64] : kb[d - 64];
    Kd[(((size_t)b * kNKV + g) * kS + s) * kHD + d] = x * cr[d] + o * sr[d];
    Vd[(((size_t)b * kNKV + g) * kS + s) * kHD + d] = kb[kHD + d];
  }
}

// ---------------------------------------------------------------------------
// Flash attention: one wave per 16-row query tile, causal, online softmax.
// QK^T and PV both on v_wmma_f32_16x16x4_f32. Out layout (B, S, NH, HD).
// K/V stream through L2 (33MB total, fits the 192MB L2 many times over).
// ---------------------------------------------------------------------------
__global__ __launch_bounds__(128) void flash_attn(
    const float* __restrict__ Q, const float* __restrict__ Kd,
    const float* __restrict__ Vd, float* __restrict__ O)
{
  __shared__ float lp[4][16][17];   // per-wave P bounce (C-layout -> A-layout)

  const int tid  = threadIdx.x;
  const int lane = tid & 31;
  const int wave = tid >> 5;
  const int half = lane >> 4;
  const int klo  = half << 1;
  const int col  = lane & 15;

  const int bh = blockIdx.y;
  const int b  = bh / kNH;
  const int h  = bh % kNH;
  const int g  = h / (kNH / kNKV);
  const int q0 = blockIdx.x * 64 + wave * 16;

  const float scale = 0.08838834764831845f;  // 1/sqrt(128)

  const float* Qb = Q  + (((size_t)b * kNH  + h) * kS) * kHD;
  const float* Kb = Kd + (((size_t)b * kNKV + g) * kS) * kHD;
  const float* Vb = Vd + (((size_t)b * kNKV + g) * kS) * kHD;

  // Preload this wave's Q A-fragments (row = lane%16, K chunks of 4)
  v2f qf[32];
  {
    const float* qrow = Qb + (size_t)(q0 + col) * kHD;
#pragma unroll
    for (int kt = 0; kt < 32; ++kt) {
      qf[kt].x = qrow[kt * 4 + klo];
      qf[kt].y = qrow[kt * 4 + klo + 1];
    }
  }

  v8f acc[8] = {};
  float mrow[8], lrow[8];
#pragma unroll
  for (int i = 0; i < 8; ++i) { mrow[i] = -1.0e30f; lrow[i] = 0.0f; }

  const int ntiles = (q0 >> 4) + 1;   // causal: only KV tiles up to the diagonal
  for (int t = 0; t < ntiles; ++t) {
    const int kv0 = t * 16;

    // S = Q @ K^T (16x16 tile, K-dim = HD = 128 -> 32 WMMAs)
    v8f s = {};
#pragma unroll
    for (int kt = 0; kt < 32; ++kt) {
      v2f kf;
      const float* krow = Kb + (size_t)(kv0 + col) * kHD + kt * 4 + klo;
      kf.x = krow[0]; kf.y = krow[1];
      s = __builtin_amdgcn_wmma_f32_16x16x4_f32(false, qf[kt], false, kf, (short)0, s, false, false);
    }

    // Scale + causal mask + online softmax (row reduce in 16-lane groups)
    float p[8];
#pragma unroll
    for (int i = 0; i < 8; ++i) {
      int row = q0 + i + 8 * half;
      int c   = kv0 + col;
      float sv = s[i] * scale + ((c <= row) ? 0.0f : -1000000000.0f);
      float rm = sv;
      rm = fmaxf(rm, __shfl_xor(rm, 1, 16));
      rm = fmaxf(rm, __shfl_xor(rm, 2, 16));
      rm = fmaxf(rm, __shfl_xor(rm, 4, 16));
      rm = fmaxf(rm, __shfl_xor(rm, 8, 16));
      float mnew = fmaxf(mrow[i], rm);
      float corr = __expf(mrow[i] - mnew);
      float pv   = __expf(sv - mnew);
      float rs = pv;
      rs += __shfl_xor(rs, 1, 16);
      rs += __shfl_xor(rs, 2, 16);
      rs += __shfl_xor(rs, 4, 16);
      rs += __shfl_xor(rs, 8, 16);
      lrow[i] = lrow[i] * corr + rs;
      mrow[i] = mnew;
      p[i] = pv;
#pragma unroll
      for (int dt = 0; dt < 8; ++dt) acc[dt][i] *= corr;
    }

    // P: C-layout -> A-layout through per-wave LDS
#pragma unroll
    for (int i = 0; i < 8; ++i) lp[wave][i + 8 * half][col] = p[i];
    asm volatile("s_wait_dscnt 0x0" ::: "memory");
    v2f pa[4];
#pragma unroll
    for (int kc = 0; kc < 4; ++kc) {
      pa[kc].x = lp[wave][col][kc * 4 + klo];
      pa[kc].y = lp[wave][col][kc * 4 + klo + 1];
    }

    // acc += P @ V  (8 d-tiles x 4 K-chunks)
#pragma unroll
    for (int dt = 0; dt < 8; ++dt) {
#pragma unroll
      for (int kc = 0; kc < 4; ++kc) {
        v2f vf;
        const float* vrow = Vb + (size_t)(kv0 + kc * 4 + klo) * kHD + dt * 16 + col;
        vf.x = vrow[0];
        vf.y = vrow[kHD];
        acc[dt] = __builtin_amdgcn_wmma_f32_16x16x4_f32(false, pa[kc], false, vf, (short)0, acc[dt], false, false);
      }
    }
  }

  // Normalize and write (B, S, NH, HD) so the output GEMM reads flat rows
#pragma unroll
  for (int i = 0; i < 8; ++i) {
    int row = q0 + i + 8 * half;
    float inv = 1.0f / lrow[i];
    float* orow = O + (((size_t)b * kS + row) * kNH + h) * kHD;
#pragma unroll
    for (int dt = 0; dt < 8; ++dt) orow[dt * 16 + col] = acc[dt][i] * inv;
  }
}

// ---------------------------------------------------------------------------
extern "C" void kernel_launch(void* const* d_in, const int* in_sizes, int n_in,
                              void* d_out, int out_size, void* d_ws, size_t ws_size,
                              hipStream_t stream) {
  (void)in_sizes; (void)n_in; (void)out_size; (void)ws_size;
  const float* hs   = (const float*)d_in[0];
  // d_in[1] attention_mask: causal mask is applied analytically in flash_attn
  const float* cosT = (const float*)d_in[2];
  const float* sinT = (const float*)d_in[3];
  const float* Wq   = (const float*)d_in[4];
  const float* Wkv  = (const float*)d_in[5];
  const float* Wd   = (const float*)d_in[6];
  const float* bd   = (const float*)d_in[7];
  float* out = (float*)d_out;

  float* ws = (float*)d_ws;
  size_t off = 0;
  float* qproj  = ws + off; off += (size_t)kB * kS * kHID;            // 16.8M
  float* kvproj = ws + off; off += (size_t)kB * kS * 2 * kNKV * kHD;  //  8.4M
  float* Qr     = ws + off; off += (size_t)kB * kNH  * kS * kHD;      // 16.8M
  float* Kr     = ws + off; off += (size_t)kB * kNKV * kS * kHD;      //  4.2M
  float* Vr     = ws + off; off += (size_t)kB * kNKV * kS * kHD;      //  4.2M
  float* attn   = qproj;  // qproj dead after rope_split; reuse for attn output

  const int M = kB * kS;  // 4096
  dim3 blk(128);

  wmma_gemm_f32<<<dim3(kHID / 64, M / 64), blk, 0, stream>>>(
      hs, Wq, nullptr, qproj, M, kHID, kHID);
  wmma_gemm_f32<<<dim3((2 * kNKV * kHD) / 64, M / 64), blk, 0, stream>>>(
      hs, Wkv, nullptr, kvproj, M, 2 * kNKV * kHD, kHID);
  rope_split<<<dim3(kB * kS), dim3(256), 0, stream>>>(
      qproj, kvproj, cosT, sinT, Qr, Kr, Vr);
  flash_attn<<<dim3(kS / 64, kB * kNH), blk, 0, stream>>>(Qr, Kr, Vr, attn);
  wmma_gemm_f32<<<dim3(kHID / 64, M / 64), blk, 0, stream>>>(
      attn, Wd, bd, out, M, kHID, kHID);
}